// ZeroShotMultimodalLSM_8761733284465
// MI455X (gfx1250) — compile-verified
//
#include <hip/hip_runtime.h>

// Problem dims (from reference)
#define T_STEPS 1024
#define BATCH   16
#define NNEUR   2048
#define MCOLS   512

typedef __attribute__((ext_vector_type(2))) float v2f;
typedef __attribute__((ext_vector_type(8))) float v8f;

// ---------------------------------------------------------------------------
// Kernel 1: LIF neuron scan. One thread per (b,n) neuron, serial over T.
// Memory-bound: 402 MB of once-read streaming input -> non-temporal loads,
// coalesced across n, 1024 wave32s in flight.
// ---------------------------------------------------------------------------
__global__ __launch_bounds__(256) void lif_scan_kernel(
    const float* __restrict__ ext,
    const float* __restrict__ inz,
    const float* __restrict__ vnz,
    float* __restrict__ rates)
{
    const int idx = blockIdx.x * 256 + threadIdx.x;   // [0, B*N)
    const int BN  = BATCH * NNEUR;

    const float DT        = 1e-7f;
    const float CMEM      = 1e-12f;
    const float G_LEAK    = 1e-9f;
    const float VTH       = 0.7f;     // VTH * GAIN, GAIN = 1
    const float VRESET    = 0.0f;
    const float REFRAC    = 1e-6f;
    const float DECAY     = 1.0f - (1e-7f / 1e-4f);   // 1 - DT/TAU_ADAPT
    const float ADAPT_INC = 5e-11f;

    float V = 0.0f, A = 0.0f, last_spike = -1e9f;
    int spikes = 0;

#pragma unroll 4
    for (int t = 0; t < T_STEPS; ++t) {
        const int off = t * BN + idx;
        const float e  = __builtin_nontemporal_load(ext + off);
        const float in = __builtin_nontemporal_load(inz + off);
        const float vn = __builtin_nontemporal_load(vnz + off);
        const float tf = (float)t * DT;

        const bool  active  = (tf - last_spike) >= REFRAC;
        const float total_I = e - A + in;
        const float Vn = V + DT * (total_I - G_LEAK * V) / CMEM + vn;  // + OFFSET(0)
        const float An = A * DECAY;

        const float V1 = active ? Vn : V;
        const float A1 = active ? An : A;
        const bool  spike = active && (V1 >= VTH);

        V          = spike ? VRESET            : V1;
        A          = spike ? (A1 + ADAPT_INC)  : A1;
        last_spike = spike ? tf                : last_spike;
        spikes    += spike ? 1 : 0;
    }
    rates[idx] = (float)spikes * (1.0f / (float)T_STEPS);
}

// ---------------------------------------------------------------------------
// Kernel 2: noisy-conductance crossbar readout via V_WMMA_F32_16X16X4_F32.
// out[16,512] = rates[16,2048] @ (max(1e-8, G*(1+0.1118*nz)) * 0.1)
// One block per 16-column tile (32 blocks). 8 waves split K=2048; each wave
// does 64 WMMAs with 2 interleaved accumulators, partials reduced in LDS.
// ---------------------------------------------------------------------------
__global__ __launch_bounds__(256) void crossbar_wmma_kernel(
    const float* __restrict__ rates,
    const float* __restrict__ G,
    const float* __restrict__ cnz,
    float* __restrict__ out)
{
    __shared__ float lds[8 * 256];   // 8 partial 16x16 f32 tiles

    const int tid  = threadIdx.x;
    const int wave = tid >> 5;        // 0..7
    const int lane = tid & 31;        // 0..31
    const int half = lane >> 4;       // 0 | 1
    const int l    = lane & 15;       // 0..15
    const int col  = blockIdx.x * 16 + l;

    const int KCHUNK = NNEUR / 8;     // 256 per wave
    const int kbase  = wave * KCHUNK;

    v8f c0 = {};
    v8f c1 = {};

    for (int k0 = kbase; k0 < kbase + KCHUNK; k0 += 8) {
        // ---- first k-quad (K = k0 .. k0+3) ----
        {
            const int ka = k0 + 2 * half;
            v2f a = *(const v2f*)(rates + l * NNEUR + ka);      // A[M=l][ka, ka+1]
            const float g0 = G  [ka * MCOLS + col];
            const float n0 = cnz[ka * MCOLS + col];
            const float g1 = G  [(ka + 1) * MCOLS + col];
            const float n1 = cnz[(ka + 1) * MCOLS + col];
            v2f b;
            b.x = fmaxf(1e-8f, g0 * (1.0f + 0.1118f * n0)) * 0.1f;
            b.y = fmaxf(1e-8f, g1 * (1.0f + 0.1118f * n1)) * 0.1f;
            c0 = __builtin_amdgcn_wmma_f32_16x16x4_f32(
                     false, a, false, b, (short)0, c0, false, false);
        }
        // ---- second k-quad (K = k0+4 .. k0+7), independent accumulator ----
        {
            const int ka = k0 + 4 + 2 * half;
            v2f a = *(const v2f*)(rates + l * NNEUR + ka);
            const float g0 = G  [ka * MCOLS + col];
            const float n0 = cnz[ka * MCOLS + col];
            const float g1 = G  [(ka + 1) * MCOLS + col];
            const float n1 = cnz[(ka + 1) * MCOLS + col];
            v2f b;
            b.x = fmaxf(1e-8f, g0 * (1.0f + 0.1118f * n0)) * 0.1f;
            b.y = fmaxf(1e-8f, g1 * (1.0f + 0.1118f * n1)) * 0.1f;
            c1 = __builtin_amdgcn_wmma_f32_16x16x4_f32(
                     false, a, false, b, (short)0, c1, false, false);
        }
    }

    // Stash partial tile: element (row, col_in_tile) at lds[wave*256 + row*16 + l]
#pragma unroll
    for (int r = 0; r < 8; ++r) {
        const int row = r + 8 * half;
        lds[wave * 256 + row * 16 + l] = c0[r] + c1[r];
    }
    __syncthreads();

    // Reduce 8 partials: each of the 256 threads owns one output element.
    const int row = tid >> 4;         // 0..15
    const int cc  = tid & 15;         // 0..15
    float s = 0.0f;
#pragma unroll
    for (int w = 0; w < 8; ++w) s += lds[w * 256 + row * 16 + cc];
    out[row * MCOLS + blockIdx.x * 16 + cc] = s;
}

// ---------------------------------------------------------------------------
extern "C" void kernel_launch(void* const* d_in, const int* in_sizes, int n_in,
                              void* d_out, int out_size, void* d_ws, size_t ws_size,
                              hipStream_t stream) {
    const float* ext = (const float*)d_in[0];   // [T,B,N]
    const float* inz = (const float*)d_in[1];   // [T,B,N]
    const float* vnz = (const float*)d_in[2];   // [T,B,N]
    const float* G   = (const float*)d_in[3];   // [N,M]
    const float* cnz = (const float*)d_in[4];   // [N,M]
    float* rates = (float*)d_ws;                // [B,N] = 128 KB scratch
    float* out   = (float*)d_out;               // [B,M]

    lif_scan_kernel<<<(BATCH * NNEUR) / 256, 256, 0, stream>>>(ext, inz, vnz, rates);
    crossbar_wmma_kernel<<<MCOLS / 16, 256, 0, stream>>>(rates, G, cnz, out);
}